// GraphEncoderGINE_69303592288943
// MI455X (gfx1250) — compile-verified
//
#include <hip/hip_runtime.h>
#include <hip/hip_bf16.h>

typedef float v2f __attribute__((ext_vector_type(2)));
typedef float v8f __attribute__((ext_vector_type(8)));

#define N_NODES  20000
#define N_EDGES  320000
#define X_DIM    128
#define EDGE_DIM 64
#define HIDDEN   256
#define OUT_DIM  256
#define N_LAYERS 3
#define N_GRAPHS 64

__device__ __forceinline__ v8f wmma4(v2f a, v2f b, v8f c) {
    // V_WMMA_F32_16X16X4_F32: (neg_a, A, neg_b, B, c_mod, C, reuse_a, reuse_b)
    return __builtin_amdgcn_wmma_f32_16x16x4_f32(false, a, false, b, (short)0, c,
                                                 false, false);
}

// ---------------------------------------------------------------------------
// Generic fused GEMM: D[M,N] = act( (A0 (+A1)) @ W[K,N] + bias ), one 16x64
// output tile per wave32.  K multiple of 4, N multiple of 64.
// HAS_A1 / RELU are compile-time so the K-loop is a single branch-free block.
// ---------------------------------------------------------------------------
template <bool HAS_A1, bool RELU>
__global__ __launch_bounds__(256)
void gemm_bias_act(const float* __restrict__ A0, const float* __restrict__ A1,
                   const float* __restrict__ W,  const float* __restrict__ bias,
                   float* __restrict__ D, int M, int K, int N)
{
    const int lane = threadIdx.x & 31;
    const int wave = blockIdx.x * (blockDim.x >> 5) + (threadIdx.x >> 5);
    const int nbCount = N >> 6;
    const int mt = wave / nbCount;
    const int nb = wave - mt * nbCount;
    const int row0 = mt << 4;
    if (row0 >= M) return;
    const int col0 = nb << 6;

    const int aRow = row0 + (lane & 15);
    const int aR   = (aRow < M) ? aRow : (M - 1);   // clamp; stores are masked
    const int kOff = (lane >> 4) << 1;              // 0 for lanes 0-15, 2 for 16-31
    const int nc   = col0 + (lane & 15);

    // Loop-carried pointers: no per-iteration 64-bit multiplies.
    const float* ap = A0 + (size_t)aR * K + kOff;
    const float* aq = HAS_A1 ? (A1 + (size_t)aR * K + kOff) : nullptr;
    const float* wp = W + (size_t)kOff * N + nc;
    const size_t wstep = (size_t)4 * N;

    v8f c0 = {}, c1 = {}, c2 = {}, c3 = {};
#pragma unroll 2
    for (int k0 = 0; k0 < K; k0 += 4) {
        v2f a; a.x = ap[0]; a.y = ap[1];
        if (HAS_A1) { a.x += aq[0]; a.y += aq[1]; aq += 4; }
        v2f b0, b1, b2, b3;
        b0.x = wp[0];  b0.y = wp[N];
        b1.x = wp[16]; b1.y = wp[N + 16];
        b2.x = wp[32]; b2.y = wp[N + 32];
        b3.x = wp[48]; b3.y = wp[N + 48];
        ap += 4;
        wp += wstep;
        c0 = wmma4(a, b0, c0);
        c1 = wmma4(a, b1, c1);
        c2 = wmma4(a, b2, c2);
        c3 = wmma4(a, b3, c3);
    }

    const int rAdd = (lane >> 4) << 3;   // upper lane-half holds rows +8
    const int cb   = col0 + (lane & 15);
    const float bi0 = bias[cb],      bi1 = bias[cb + 16];
    const float bi2 = bias[cb + 32], bi3 = bias[cb + 48];
#pragma unroll
    for (int r = 0; r < 8; ++r) {
        const int row = row0 + r + rAdd;
        if (row < M) {
            float v0 = c0[r] + bi0, v1 = c1[r] + bi1;
            float v2 = c2[r] + bi2, v3 = c3[r] + bi3;
            if (RELU) {
                v0 = fmaxf(v0, 0.f); v1 = fmaxf(v1, 0.f);
                v2 = fmaxf(v2, 0.f); v3 = fmaxf(v3, 0.f);
            }
            float* dp = D + (size_t)row * N;
            dp[cb] = v0; dp[cb + 16] = v1; dp[cb + 32] = v2; dp[cb + 48] = v3;
        }
    }
}

// ---------------------------------------------------------------------------
// Fused GINE message + scatter:  per 16-edge x 64-col tile,
//   msg = relu(edge_attr@W + b + h[src]);  atomicAdd(agg[dst], msg)
// The [E,HIDDEN] message tensor is never materialized (saves ~327MB/layer of
// HBM traffic); agg (20MB) is L2-resident so the f32 atomics land in L2.
// ---------------------------------------------------------------------------
__global__ __launch_bounds__(256)
void edge_msg_scatter(const float* __restrict__ EA,   // [E, EDGE_DIM]
                      const float* __restrict__ W,    // [EDGE_DIM, HIDDEN]
                      const float* __restrict__ bias, // [HIDDEN]
                      const float* __restrict__ H,    // [N_NODES, HIDDEN]
                      const int*   __restrict__ src,
                      const int*   __restrict__ dst,
                      float* __restrict__ AGG)        // [N_NODES, HIDDEN]
{
    const int K = EDGE_DIM, N = HIDDEN;
    const int lane = threadIdx.x & 31;
    const int wave = blockIdx.x * (blockDim.x >> 5) + (threadIdx.x >> 5);
    const int tile = wave >> 2;       // 16-edge tile index
    const int nb   = wave & 3;        // 64-column strip
    const int e0   = tile << 4;
    if (e0 >= N_EDGES) return;
    const int col0 = nb << 6;

    const int aRow = e0 + (lane & 15);
    const int kOff = (lane >> 4) << 1;
    const int nc   = col0 + (lane & 15);

    const float* ap = EA + (size_t)aRow * K + kOff;
    const float* wp = W + (size_t)kOff * N + nc;
    const size_t wstep = (size_t)4 * N;

    // Stream the next edge tile's A-panel toward L2 (global_prefetch_b8).
    if (e0 + 16 < N_EDGES)
        __builtin_prefetch(ap + (size_t)16 * K, 0, 1);

    v8f c0 = {}, c1 = {}, c2 = {}, c3 = {};
#pragma unroll 4
    for (int k0 = 0; k0 < K; k0 += 4) {
        v2f a; a.x = ap[0]; a.y = ap[1];
        v2f b0, b1, b2, b3;
        b0.x = wp[0];  b0.y = wp[N];
        b1.x = wp[16]; b1.y = wp[N + 16];
        b2.x = wp[32]; b2.y = wp[N + 32];
        b3.x = wp[48]; b3.y = wp[N + 48];
        ap += 4;
        wp += wstep;
        c0 = wmma4(a, b0, c0);
        c1 = wmma4(a, b1, c1);
        c2 = wmma4(a, b2, c2);
        c3 = wmma4(a, b3, c3);
    }

    const int rAdd = (lane >> 4) << 3;
    const int cb   = col0 + (lane & 15);
    const float bi0 = bias[cb],      bi1 = bias[cb + 16];
    const float bi2 = bias[cb + 32], bi3 = bias[cb + 48];
#pragma unroll
    for (int r = 0; r < 8; ++r) {
        const int e = e0 + r + rAdd;             // N_EDGES % 16 == 0
        const int s = src[e], d = dst[e];
        const float* hp = H + (size_t)s * N;
        float* apg = AGG + (size_t)d * N;
        const float v0 = fmaxf(c0[r] + bi0 + hp[cb],      0.f);
        const float v1 = fmaxf(c1[r] + bi1 + hp[cb + 16], 0.f);
        const float v2 = fmaxf(c2[r] + bi2 + hp[cb + 32], 0.f);
        const float v3 = fmaxf(c3[r] + bi3 + hp[cb + 48], 0.f);
        unsafeAtomicAdd(&apg[cb],      v0);
        unsafeAtomicAdd(&apg[cb + 16], v1);
        unsafeAtomicAdd(&apg[cb + 32], v2);
        unsafeAtomicAdd(&apg[cb + 48], v3);
    }
}

__global__ void fill_zero(float* __restrict__ p, long long n) {
    long long i = (long long)blockIdx.x * blockDim.x + threadIdx.x;
    if (i < n) p[i] = 0.f;
}

__global__ void pool_scatter(const float* __restrict__ H,
                             const int* __restrict__ batch,
                             float* __restrict__ sums, float* __restrict__ cnts)
{
    const int i = blockIdx.x;          // node
    const int c = threadIdx.x;         // channel
    const int b = batch[i];
    unsafeAtomicAdd(&sums[(size_t)b * HIDDEN + c], H[(size_t)i * HIDDEN + c]);
    if (c == 0) unsafeAtomicAdd(&cnts[b], 1.0f);
}

__global__ void pool_finalize(float* __restrict__ sums, const float* __restrict__ cnts) {
    const int b = blockIdx.x, c = threadIdx.x;
    sums[(size_t)b * HIDDEN + c] /= fmaxf(cnts[b], 1.0f);
}

__global__ void l2_normalize(const float* __restrict__ G, float* __restrict__ OUT) {
    __shared__ float sh[OUT_DIM];
    const int b = blockIdx.x, c = threadIdx.x;
    const float v = G[(size_t)b * OUT_DIM + c];
    sh[c] = v * v;
    __syncthreads();
    for (int s = OUT_DIM / 2; s > 0; s >>= 1) {
        if (c < s) sh[c] += sh[c + s];
        __syncthreads();
    }
    const float nrm = fmaxf(sqrtf(sh[0]), 1e-12f);
    OUT[(size_t)b * OUT_DIM + c] = v / nrm;
}

static inline int wave_blocks(int waves) { return (waves + 7) / 8; }   // 8 waves/block

extern "C" void kernel_launch(void* const* d_in, const int* in_sizes, int n_in,
                              void* d_out, int out_size, void* d_ws, size_t ws_size,
                              hipStream_t stream) {
    const float* x       = (const float*)d_in[0];
    const int*   ei      = (const int*)  d_in[1];
    const float* ea      = (const float*)d_in[2];
    const int*   batch   = (const int*)  d_in[3];
    const float* xproj_W = (const float*)d_in[4];
    const float* xproj_b = (const float*)d_in[5];
    const float* edge_W  = (const float*)d_in[6];
    const float* edge_b  = (const float*)d_in[7];
    const float* mlp_W1  = (const float*)d_in[8];
    const float* mlp_b1  = (const float*)d_in[9];
    const float* mlp_W2  = (const float*)d_in[10];
    const float* mlp_b2  = (const float*)d_in[11];
    const float* out_W1  = (const float*)d_in[12];
    const float* out_b1  = (const float*)d_in[13];
    const float* out_W2  = (const float*)d_in[14];
    const float* out_b2  = (const float*)d_in[15];
    const int* src = ei;
    const int* dst = ei + N_EDGES;

    const size_t NH = (size_t)N_NODES * HIDDEN;
    float* ws   = (float*)d_ws;
    float* h    = ws;               // [N_NODES, HIDDEN]
    float* agg  = h   + NH;         // [N_NODES, HIDDEN]
    float* tmp  = agg + NH;         // [N_NODES, HIDDEN]
    float* gsum = tmp + NH;         // [N_GRAPHS, HIDDEN]
    float* gcnt = gsum + (size_t)N_GRAPHS * HIDDEN;  // [N_GRAPHS]
    float* g1   = gcnt + N_GRAPHS;  // [N_GRAPHS, HIDDEN]
    float* g2   = g1   + (size_t)N_GRAPHS * HIDDEN;  // [N_GRAPHS, OUT_DIM]

    const int nodeWaves = ((N_NODES + 15) / 16) * (HIDDEN / 64);
    const int edgeWaves = (N_EDGES / 16) * 4;

    // h = x @ xproj_W + xproj_b
    gemm_bias_act<false, false><<<wave_blocks(nodeWaves), 256, 0, stream>>>(
        x, nullptr, xproj_W, xproj_b, h, N_NODES, X_DIM, HIDDEN);

    for (int l = 0; l < N_LAYERS; ++l) {
        const long long n = (long long)NH;
        fill_zero<<<(unsigned)((n + 255) / 256), 256, 0, stream>>>(agg, n);
        edge_msg_scatter<<<wave_blocks(edgeWaves), 256, 0, stream>>>(
            ea, edge_W + (size_t)l * EDGE_DIM * HIDDEN, edge_b + (size_t)l * HIDDEN,
            h, src, dst, agg);
        // tmp = relu((h+agg) @ W1 + b1)
        gemm_bias_act<true, true><<<wave_blocks(nodeWaves), 256, 0, stream>>>(
            h, agg, mlp_W1 + (size_t)l * HIDDEN * HIDDEN, mlp_b1 + (size_t)l * HIDDEN,
            tmp, N_NODES, HIDDEN, HIDDEN);
        // h = relu(tmp @ W2 + b2)
        gemm_bias_act<false, true><<<wave_blocks(nodeWaves), 256, 0, stream>>>(
            tmp, nullptr, mlp_W2 + (size_t)l * HIDDEN * HIDDEN, mlp_b2 + (size_t)l * HIDDEN,
            h, N_NODES, HIDDEN, HIDDEN);
    }

    // global mean pool
    fill_zero<<<(N_GRAPHS * HIDDEN + N_GRAPHS + 255) / 256, 256, 0, stream>>>(
        gsum, (long long)(N_GRAPHS * HIDDEN + N_GRAPHS));
    pool_scatter<<<N_NODES, HIDDEN, 0, stream>>>(h, batch, gsum, gcnt);
    pool_finalize<<<N_GRAPHS, HIDDEN, 0, stream>>>(gsum, gcnt);

    const int outWaves = ((N_GRAPHS + 15) / 16) * (HIDDEN / 64);
    gemm_bias_act<false, true><<<wave_blocks(outWaves), 256, 0, stream>>>(
        gsum, nullptr, out_W1, out_b1, g1, N_GRAPHS, HIDDEN, HIDDEN);
    gemm_bias_act<false, false><<<wave_blocks(outWaves), 256, 0, stream>>>(
        g1, nullptr, out_W2, out_b2, g2, N_GRAPHS, HIDDEN, OUT_DIM);

    l2_normalize<<<N_GRAPHS, OUT_DIM, 0, stream>>>(g2, (float*)d_out);
}